// EfficientAttention_70274254897248
// MI455X (gfx1250) — compile-verified
//
#include <hip/hip_runtime.h>

#define ALPHA  0.2f
#define NBATCH 2
#define HWSZ   4096
#define TTOK   262144
#define NCHUNK 32

typedef __attribute__((ext_vector_type(2))) float v2f;
typedef __attribute__((ext_vector_type(8))) float v8f;

__device__ __forceinline__ float lrelu(float v) { return v >= 0.0f ? v : ALPHA * v; }

__device__ __forceinline__ v8f zero8() {
    v8f z = {0.f,0.f,0.f,0.f,0.f,0.f,0.f,0.f};
    return z;
}

// Wave-level 64x64x64 f32 GEMM using V_WMMA_F32_16X16X4_F32.
// A (tokens x 64) in LDS row-major stride 65, B (64 x 64) in LDS stride 65.
// Each wave computes rows [c0, c0+16), all 64 columns -> acc[4] (4 N-tiles of 16).
__device__ __forceinline__ void gemm64_wmma(const float* __restrict__ As,
                                            const float* __restrict__ Bs,
                                            int c0, int lane, v8f acc[4])
{
    const int m  = lane & 15;          // M (A) / N (B) index within tile
    const int kh = (lane >> 4) << 1;   // K sub-offset: lanes 0-15 -> K{0,1}, 16-31 -> K{2,3}
    #pragma unroll
    for (int k0 = 0; k0 < 64; k0 += 4) {
        const int ka = k0 + kh;
        v2f a;
        a.x = As[(c0 + m) * 65 + ka];
        a.y = As[(c0 + m) * 65 + ka + 1];
        #pragma unroll
        for (int nt = 0; nt < 4; ++nt) {
            v2f b;
            b.x = Bs[ka * 65 + nt * 16 + m];
            b.y = Bs[(ka + 1) * 65 + nt * 16 + m];
            acc[nt] = __builtin_amdgcn_wmma_f32_16x16x4_f32(
                false, a, false, b, (short)0, acc[nt], false, false);
        }
    }
}

// ---------------------------------------------------------------------------
// Pass 1: one workgroup per (n,h,w): 64 tokens (c=0..63), feature dim d=0..63.
//  - K proj -> exp() -> running Z[64] (denominator) and S[a][kk][vv] (ctx numerator)
//  - V proj -> folded into S (ek x V outer products, reduced in-registers + LDS atomics)
//  - Q proj -> per-head softmax via lane shuffles -> stored to global (only big output)
// ---------------------------------------------------------------------------
__global__ __launch_bounds__(128) void k_proj(
    const float* __restrict__ x,
    const float* __restrict__ Wk, const float* __restrict__ sk, const float* __restrict__ bk,
    const float* __restrict__ Wq, const float* __restrict__ sq, const float* __restrict__ bq,
    const float* __restrict__ Wv, const float* __restrict__ sv, const float* __restrict__ bv,
    float* __restrict__ Qout, float* __restrict__ Part)
{
    __shared__ float Xs[64 * 65];   // X[c][d] = x[n,d,h,w,c]
    __shared__ float Wb[64 * 65];   // current weight matrix (reloaded per projection)
    __shared__ float Es[64 * 65];   // exp(keys) tile
    __shared__ float Sl[512];       // block-local ctx numerator S[a][kk][vv]
    __shared__ float Zl[64];        // block-local denominator per key channel

    const int tid  = threadIdx.x;
    const int wave = tid >> 5;
    const int lane = tid & 31;
    const int bid  = blockIdx.x;
    const int n    = bid >> 12;
    const int hw   = bid & 4095;

    // load & transpose x tile: rows x[n,d,h,w,0:64] are 256B contiguous
    #pragma unroll
    for (int r = 0; r < 16; ++r) {
        const int d = wave * 16 + r;
        const float2* row = (const float2*)(x + (((size_t)(n * 64 + d)) * HWSZ + hw) * 64);
        const float2 v = row[lane];
        Xs[(2 * lane) * 65 + d]     = v.x;
        Xs[(2 * lane + 1) * 65 + d] = v.y;
    }
    for (int i = tid; i < 512; i += 128) Sl[i] = 0.0f;
    if (tid < 64) Zl[tid] = 0.0f;
    for (int i = tid; i < 4096; i += 128) Wb[(i >> 6) * 65 + (i & 63)] = Wk[i];
    __syncthreads();

    const int c0    = wave * 16;
    const int m     = lane & 15;
    const int khalf = lane >> 4;

    // ---- keys -> exp tile + Z
    {
        v8f acc[4];
        #pragma unroll
        for (int nt = 0; nt < 4; ++nt) acc[nt] = zero8();
        gemm64_wmma(Xs, Wb, c0, lane, acc);
        #pragma unroll
        for (int nt = 0; nt < 4; ++nt) {
            const int e = nt * 16 + m;
            const float sc = sk[e], bi = bk[e];
            float zs = 0.0f;
            #pragma unroll
            for (int r = 0; r < 8; ++r) {
                const int tok = c0 + khalf * 8 + r;
                const float ek = __expf(lrelu(acc[nt][r] * sc + bi));
                Es[tok * 65 + e] = ek;
                zs += ek;
            }
            zs += __shfl_xor(zs, 16);
            if (khalf == 0) atomicAdd(&Zl[e], zs);
        }
    }
    __syncthreads();
    for (int i = tid; i < 4096; i += 128) Wb[(i >> 6) * 65 + (i & 63)] = Wv[i];
    __syncthreads();

    // ---- values -> S[a][kk][vv] += ek[t, 8a+kk] * V[t, 8a+vv]
    {
        v8f acc[4];
        #pragma unroll
        for (int nt = 0; nt < 4; ++nt) acc[nt] = zero8();
        gemm64_wmma(Xs, Wb, c0, lane, acc);
        #pragma unroll
        for (int nt = 0; nt < 4; ++nt) {
            const int e = nt * 16 + m;
            const int a = e >> 3, vv = e & 7;
            const float sc = sv[e], bi = bv[e];
            float vreg[8];
            #pragma unroll
            for (int r = 0; r < 8; ++r) vreg[r] = lrelu(acc[nt][r] * sc + bi);
            float ssum[8];
            #pragma unroll
            for (int kk = 0; kk < 8; ++kk) ssum[kk] = 0.0f;
            #pragma unroll
            for (int r = 0; r < 8; ++r) {
                const int tok = c0 + khalf * 8 + r;
                const float* er = &Es[tok * 65 + a * 8];
                #pragma unroll
                for (int kk = 0; kk < 8; ++kk) ssum[kk] += er[kk] * vreg[r];
            }
            #pragma unroll
            for (int kk = 0; kk < 8; ++kk) {
                float sred = ssum[kk] + __shfl_xor(ssum[kk], 16);
                if (khalf == 0) atomicAdd(&Sl[a * 64 + kk * 8 + vv], sred);
            }
        }
    }
    __syncthreads();   // Sl/Zl complete; Wb free for reuse
    for (int i = tid; i < 4096; i += 128) Wb[(i >> 6) * 65 + (i & 63)] = Wq[i];
    __syncthreads();

    // ---- queries -> softmax over 8 head channels (lanes m%8) -> global store
    {
        v8f acc[4];
        #pragma unroll
        for (int nt = 0; nt < 4; ++nt) acc[nt] = zero8();
        gemm64_wmma(Xs, Wb, c0, lane, acc);
        #pragma unroll
        for (int nt = 0; nt < 4; ++nt) {
            const int e = nt * 16 + m;
            const float sc = sq[e], bi = bq[e];
            #pragma unroll
            for (int r = 0; r < 8; ++r) {
                const int tok = c0 + khalf * 8 + r;
                const float q = lrelu(acc[nt][r] * sc + bi);
                float mx = q;
                mx = fmaxf(mx, __shfl_xor(mx, 1));
                mx = fmaxf(mx, __shfl_xor(mx, 2));
                mx = fmaxf(mx, __shfl_xor(mx, 4));
                const float ex = __expf(q - mx);
                float sm = ex;
                sm += __shfl_xor(sm, 1);
                sm += __shfl_xor(sm, 2);
                sm += __shfl_xor(sm, 4);
                Qout[((size_t)n * TTOK + (size_t)tok * HWSZ + hw) * 64 + e] = ex / sm;
            }
        }
    }

    // ---- flush block partials into chunked global accumulators
    {
        float* base = Part + ((size_t)(n * NCHUNK + (bid & (NCHUNK - 1)))) * 576;
        #pragma unroll
        for (int j = 0; j < 4; ++j)
            atomicAdd(&base[tid * 4 + j], Sl[tid * 4 + j]);
        if (tid < 64) atomicAdd(&base[512 + tid], Zl[tid]);
    }
}

__global__ void k_zero(float* __restrict__ p, int nfl)
{
    const int i = blockIdx.x * 256 + threadIdx.x;
    if (i < nfl) p[i] = 0.0f;
}

// ctx[n,a,kk,vv] = S/Z  (reduce the NCHUNK chunk copies)
__global__ void k_ctx(const float* __restrict__ Part, float* __restrict__ ctx)
{
    const int i = blockIdx.x * 256 + threadIdx.x;
    if (i >= NBATCH * 512) return;
    const int n = i >> 9, rem = i & 511;
    const int kc = rem >> 3;                    // a*8+kk
    float s = 0.0f, z = 0.0f;
    for (int c = 0; c < NCHUNK; ++c) {
        const float* base = Part + ((size_t)(n * NCHUNK + c)) * 576;
        s += base[rem];
        z += base[512 + kc];
    }
    ctx[i] = s / z;
}

// ---------------------------------------------------------------------------
// Pass 2: one workgroup per (n,h,w).
//  agg[c][8a+v] = sum_{v'} ctx[a, c>>3, v'] * Q[(c&7)*2^15 + h*2^9 + w*8 + v, 8a+v']
//  rep = LeakyReLU(agg*Wr*sr+br)  (WMMA, written back in-place)
//  out[n,d,h,w,c] = x[n,d,h,w,c] + rep[c][d]
// ---------------------------------------------------------------------------
__global__ __launch_bounds__(128) void k_att(
    const float* __restrict__ x, const float* __restrict__ Qbuf, const float* __restrict__ ctx,
    const float* __restrict__ Wr, const float* __restrict__ sr, const float* __restrict__ br,
    float* __restrict__ out)
{
    __shared__ float ctxs[512];
    __shared__ float Qls[4096];      // [clow][v][e], 8 contiguous 2KB blocks
    __shared__ float Ag[64 * 65];    // agg, then rep in place
    __shared__ float Ws[64 * 65];

    const int tid  = threadIdx.x;
    const int wave = tid >> 5, lane = tid & 31;
    const int bid  = blockIdx.x;
    const int n    = bid >> 12, hw = bid & 4095;
    const int h    = hw >> 6,   w  = hw & 63;

    for (int i = tid; i < 512; i += 128) ctxs[i] = ctx[n * 512 + i];
    for (int i = tid; i < 4096; i += 128) {
        const int clow = i >> 9, rem = i & 511;
        Qls[i] = Qbuf[((size_t)n * TTOK + (size_t)clow * 32768 + h * 512 + w * 8) * 64 + rem];
    }
    for (int i = tid; i < 4096; i += 128) Ws[(i >> 6) * 65 + (i & 63)] = Wr[i];
    __syncthreads();

    #pragma unroll 4
    for (int j = 0; j < 32; ++j) {
        const int o = j * 128 + tid;
        const int c = o >> 6, e = o & 63;
        const int a = e >> 3, v = e & 7;
        const int k = c >> 3, clow = c & 7;
        const float* cx = &ctxs[a * 64 + k * 8];
        const float* qr = &Qls[clow * 512 + v * 64 + a * 8];
        float s = 0.0f;
        #pragma unroll
        for (int vp = 0; vp < 8; ++vp) s += cx[vp] * qr[vp];
        Ag[c * 65 + e] = s;
    }
    __syncthreads();

    const int c0 = wave * 16, m = lane & 15, khalf = lane >> 4;
    v8f acc[4];
    #pragma unroll
    for (int nt = 0; nt < 4; ++nt) acc[nt] = zero8();
    gemm64_wmma(Ag, Ws, c0, lane, acc);
    #pragma unroll
    for (int nt = 0; nt < 4; ++nt) {
        const int d = nt * 16 + m;
        const float sc = sr[d], bi = br[d];
        #pragma unroll
        for (int r = 0; r < 8; ++r) {
            const int tok = c0 + khalf * 8 + r;     // in-place: wave owns these rows
            Ag[tok * 65 + d] = lrelu(acc[nt][r] * sc + bi);
        }
    }
    __syncthreads();

    {
        const int d  = tid >> 1;
        const int ch = (tid & 1) * 32;
        const size_t g = (((size_t)(n * 64 + d)) * HWSZ + hw) * 64 + ch;
        const float4* xr = (const float4*)(x + g);
        float4* orow = (float4*)(out + g);
        #pragma unroll
        for (int i = 0; i < 8; ++i) {
            float4 xv = xr[i];
            const int cc = ch + i * 4;
            xv.x += Ag[cc * 65 + d];
            xv.y += Ag[(cc + 1) * 65 + d];
            xv.z += Ag[(cc + 2) * 65 + d];
            xv.w += Ag[(cc + 3) * 65 + d];
            orow[i] = xv;
        }
    }
}

extern "C" void kernel_launch(void* const* d_in, const int* in_sizes, int n_in,
                              void* d_out, int out_size, void* d_ws, size_t ws_size,
                              hipStream_t stream)
{
    (void)in_sizes; (void)n_in; (void)out_size; (void)ws_size;
    const float* x  = (const float*)d_in[0];
    const float* Wk = (const float*)d_in[1];
    const float* sk = (const float*)d_in[2];
    const float* bk = (const float*)d_in[3];
    const float* Wq = (const float*)d_in[4];
    const float* sq = (const float*)d_in[5];
    const float* bq = (const float*)d_in[6];
    const float* Wv = (const float*)d_in[7];
    const float* sv = (const float*)d_in[8];
    const float* bv = (const float*)d_in[9];
    const float* Wr = (const float*)d_in[10];
    const float* sr = (const float*)d_in[11];
    const float* br = (const float*)d_in[12];
    float* out = (float*)d_out;

    // workspace: Q (128MB) + chunked partials + ctx  (~134.4MB total)
    float* Qbuf = (float*)d_ws;
    float* Part = Qbuf + (size_t)NBATCH * TTOK * 64;
    float* ctx  = Part + (size_t)NBATCH * NCHUNK * 576;

    const int npart = NBATCH * NCHUNK * 576;
    k_zero<<<dim3((npart + 255) / 256), dim3(256), 0, stream>>>(Part, npart);
    k_proj<<<dim3(NBATCH * HWSZ), dim3(128), 0, stream>>>(
        x, Wk, sk, bk, Wq, sq, bq, Wv, sv, bv, Qbuf, Part);
    k_ctx<<<dim3((NBATCH * 512 + 255) / 256), dim3(256), 0, stream>>>(Part, ctx);
    k_att<<<dim3(NBATCH * HWSZ), dim3(128), 0, stream>>>(x, Qbuf, ctx, Wr, sr, br, out);
}